// AALMLP_77644418777722
// MI455X (gfx1250) — compile-verified
//
#include <hip/hip_runtime.h>
#include <math.h>

// ---------------------------------------------------------------------------
// Alias-free quadratic MLP (16,64,64,96) for MI455X / gfx1250, wave32 + WMMA.
//
// Pipeline (all periodic ideal filters expressed as dense circulant matrices):
//   t1 = U @ x          per batch      (128x64)(64x6144)        [H upsample]
//   xu = U @ t1         per (b,row)    (128x64)(64x96)          [W upsample]
//   y  = act(xu W1^T+b1) W2^T + b2     fused, hidden in LDS     [MLP]
//   t2 = D @ y          per (b,row)    (64x128)(128x96)         [W lpf+decim]
//   out= D @ t2         per batch      (64x128)(128x6144)       [H lpf+decim]
// All GEMMs via V_WMMA_F32_16X16X4_F32 (fp32-faithful matrix path).
//
// Workspace layout (needs 151,060,480 bytes):
//   U: 128*64 f32 | D: 64*128 f32 | T: 16*128*64*96 f32 | X: 16*128*128*96 f32
// ---------------------------------------------------------------------------

typedef __attribute__((ext_vector_type(2))) float v2f;
typedef __attribute__((ext_vector_type(8))) float v8f;

#define PI_F 3.14159265358979323846f

// ---- build the 1-D resampling operators ------------------------------------
// U[i,h]: zero-insert-by-2 then ideal LPF (|k|<=32, edge weight 1.0), x2 gain.
// D[n,j]: ideal LPF (|k|<32 full, |k|=32 at 0.5) then take every other sample.
__global__ void gen_filters(float* __restrict__ U, float* __restrict__ D) {
  int tid = blockIdx.x * blockDim.x + threadIdx.x;
  int stride = gridDim.x * blockDim.x;
  for (int idx = tid; idx < 128 * 64; idx += stride) {
    int i = idx >> 6, h = idx & 63;
    float d = (float)(i - 2 * h);
    float s = 1.0f;
    for (int k = 1; k <= 32; ++k) s += 2.0f * cosf(PI_F * (float)k * d / 64.0f);
    U[idx] = s * (1.0f / 64.0f);
  }
  for (int idx = tid; idx < 64 * 128; idx += stride) {
    int n = idx >> 7, j = idx & 127;
    float d = (float)(2 * n - j);
    float s = 1.0f;
    for (int k = 1; k <= 31; ++k) s += 2.0f * cosf(PI_F * (float)k * d / 64.0f);
    s += cosf(PI_F * 0.5f * d);                // |k|=32 pair at weight 0.5
    D[idx] = s * (1.0f / 128.0f);
  }
}

// ---- one 16x16 fp32 WMMA tile: C[m0:,n0:] = A(MxK) * B(KxN) ----------------
// A per-lane [m][k] layout; B element B[k][n] gathered per-lane as [n][k].
__device__ __forceinline__ void gemm_tile(const float* __restrict__ A,
                                          const float* __restrict__ B,
                                          float* __restrict__ C,
                                          int m0, int n0, int K,
                                          int lda, int ldb, int ldc, int lane) {
  const int l15 = lane & 15;
  const int kh  = (lane >> 4) << 1;            // 0 or 2
  v8f acc = {};
  const float* arow = A + (size_t)(m0 + l15) * lda;
  const float* bcol = B + (n0 + l15);
  for (int k = 0; k < K; k += 4) {
    v2f a = *(const v2f*)(arow + k + kh);      // contiguous, 8B aligned
    v2f b;
    b.x = bcol[(size_t)(k + kh)     * ldb];
    b.y = bcol[(size_t)(k + kh + 1) * ldb];
    acc = __builtin_amdgcn_wmma_f32_16x16x4_f32(false, a, false, b,
                                                (short)0, acc, false, false);
  }
  const int cn = n0 + l15;
  const int mb = m0 + 8 * (lane >> 4);
#pragma unroll
  for (int r = 0; r < 8; ++r)
    C[(size_t)(mb + r) * ldc + cn] = acc[r];
}

// ---- generic batched filter GEMM: C[b] = A * B[b] --------------------------
__global__ __launch_bounds__(256)
void filt_gemm(const float* __restrict__ A, const float* __restrict__ B,
               float* __restrict__ C, int M, int N, int K,
               int lda, int ldb, int ldc, long sB, long sC) {
  const int wave = threadIdx.x >> 5;
  const int lane = threadIdx.x & 31;
  const int nTm  = M >> 4;
  const int nT   = nTm * (N >> 4);
  const int tile = blockIdx.x * (blockDim.x >> 5) + wave;
  if (tile >= nT) return;                      // wave-uniform
  const float* Bb = B + (long)blockIdx.y * sB;
  float*       Cb = C + (long)blockIdx.y * sC;
  const int mt = tile % nTm, nt = tile / nTm;
  gemm_tile(A, Bb, Cb, mt * 16, nt * 16, K, lda, ldb, ldc, lane);
}

// ---- fused MLP: y = act(x W1^T + b1) W2^T + b2, in-place over x ------------
// 32 pixels per block, 8 waves. Hidden tile (32x384) lives in LDS with
// padded stride 388 (row stride mod 64 banks = 4 -> conflict-free tr-reads).
#define LDH 388
__global__ __launch_bounds__(256)
void mlp_fused(float* __restrict__ XY,          // (P,96) in: xu, out: y
               const float* __restrict__ W1,    // (384,96) row-major
               const float* __restrict__ b1,    // (384)
               const float* __restrict__ W2,    // (96,384) row-major
               const float* __restrict__ b2,    // (96)
               const float* __restrict__ coef)  // (384,3)
{
  __shared__ float Hs[32 * LDH];                // 49,664 B
  const int wave = threadIdx.x >> 5;
  const int lane = threadIdx.x & 31;
  const int l15 = lane & 15, hi = lane >> 4, kh = hi << 1;
  const long p0 = (long)blockIdx.x * 32;
  const float* Xb = XY + p0 * 96;

  // Stage A: H = act(X * W1^T + b1)   tiles: 2(M) x 24(N)
  for (int t = wave; t < 48; t += 8) {
    const int mt = t & 1, nt = t >> 1;
    v8f acc = {};
    const float* arow = Xb + (size_t)(mt * 16 + l15) * 96;
    const float* brow = W1 + (size_t)(nt * 16 + l15) * 96;   // B[n][k] contiguous
#pragma unroll
    for (int k = 0; k < 96; k += 4) {
      v2f a = *(const v2f*)(arow + k + kh);
      v2f b = *(const v2f*)(brow + k + kh);
      acc = __builtin_amdgcn_wmma_f32_16x16x4_f32(false, a, false, b,
                                                  (short)0, acc, false, false);
    }
    const int n = nt * 16 + l15;
    const float bb = b1[n];
    const float c0 = coef[n * 3 + 0], c1 = coef[n * 3 + 1], c2 = coef[n * 3 + 2];
    const int mb = mt * 16 + 8 * hi;
#pragma unroll
    for (int r = 0; r < 8; ++r) {
      float h = acc[r] + bb;
      Hs[(mb + r) * LDH + n] = c0 + h * (c1 + c2 * h);
    }
  }
  __syncthreads();

  // Stage B: Y = H * W2^T + b2        tiles: 2(M) x 6(N)
  for (int t = wave; t < 12; t += 8) {
    const int mt = t & 1, nt = t >> 1;
    v8f acc = {};
    const float* arow = Hs + (size_t)(mt * 16 + l15) * LDH;
    const float* brow = W2 + (size_t)(nt * 16 + l15) * 384;  // B[n][k] contiguous
#pragma unroll 8
    for (int k = 0; k < 384; k += 4) {
      v2f a = *(const v2f*)(arow + k + kh);
      v2f b = *(const v2f*)(brow + k + kh);
      acc = __builtin_amdgcn_wmma_f32_16x16x4_f32(false, a, false, b,
                                                  (short)0, acc, false, false);
    }
    const int n = nt * 16 + l15;
    const float bb = b2[n];
    const int mb = mt * 16 + 8 * hi;
#pragma unroll
    for (int r = 0; r < 8; ++r)
      XY[(p0 + mb + r) * 96 + n] = acc[r] + bb;   // in-place: our rows only
  }
}

// ---------------------------------------------------------------------------
extern "C" void kernel_launch(void* const* d_in, const int* in_sizes, int n_in,
                              void* d_out, int out_size, void* d_ws, size_t ws_size,
                              hipStream_t stream) {
  const float* x    = (const float*)d_in[0];   // (16,64,64,96)
  const float* W1   = (const float*)d_in[1];   // (384,96)
  const float* b1   = (const float*)d_in[2];   // (384)
  const float* W2   = (const float*)d_in[3];   // (96,384)
  const float* b2   = (const float*)d_in[4];   // (96)
  const float* coef = (const float*)d_in[5];   // (384,3)
  float* out = (float*)d_out;                  // (16,64,64,96)

  float* U = (float*)d_ws;                     // 128x64
  float* D = U + 128 * 64;                     // 64x128
  float* T = D + 64 * 128;                     // 16*128*64*96  (t1 / t2)
  float* X = T + 12582912;                     // 16*128*128*96 (xu / y)

  gen_filters<<<32, 256, 0, stream>>>(U, D);

  // t1[b] (128 x 6144) = U (128x64) @ x[b] (64 x 6144)        [H upsample]
  filt_gemm<<<dim3(384, 16), 256, 0, stream>>>(U, x, T,
      128, 6144, 64, 64, 6144, 6144, 393216L, 786432L);

  // xu[b,i] (128 x 96) = U (128x64) @ t1[b,i] (64 x 96)       [W upsample]
  filt_gemm<<<dim3(6, 2048), 256, 0, stream>>>(U, T, X,
      128, 96, 64, 64, 96, 96, 6144L, 12288L);

  // y = MLP(xu), in place over X; 262144 pixels / 32 per block
  mlp_fused<<<8192, 256, 0, stream>>>(X, W1, b1, W2, b2, coef);

  // t2[b,i] (64 x 96) = D (64x128) @ y[b,i] (128 x 96)        [W lpf+decimate]
  filt_gemm<<<dim3(3, 2048), 256, 0, stream>>>(D, X, T,
      64, 96, 128, 128, 96, 96, 12288L, 6144L);

  // out[b] (64 x 6144) = D (64x128) @ t2[b] (128 x 6144)      [H lpf+decimate]
  filt_gemm<<<dim3(192, 16), 256, 0, stream>>>(D, T, out,
      64, 6144, 128, 128, 6144, 6144, 786432L, 393216L);
}